// mnistModel_30150670418098
// MI455X (gfx1250) — compile-verified
//
#include <hip/hip_runtime.h>
#include <hip/hip_bf16.h>
#include <stdint.h>

// ---------------------------------------------------------------------------
// IterNorm-LSTM forward on MI455X (gfx1250).
// fp32 WMMA (16x16x4) for all matrix math; TDM tensor_load_to_lds stages the
// per-group whitening matrix; Newton-Schulz runs entirely in 307.5 KB of LDS.
// ---------------------------------------------------------------------------

#define SEQ   28
#define BATCH 1024
#define NINP  28
#define NHID  400
#define C4    1600          // 4*NHID gate channels
#define NG    10            // IterNorm groups per matrix
#define NG2   20            // two matrices (x-path, h-path)
#define D     160           // channels per group
#define T_NS  10            // Newton-Schulz iterations
#define EPSV  1e-5f
#define LDP   164           // padded LDS leading dim (conflict-free column reads)

typedef __attribute__((ext_vector_type(2))) float v2f;
typedef __attribute__((ext_vector_type(4))) float v4f;
typedef __attribute__((ext_vector_type(8))) float v8f;
typedef __attribute__((ext_vector_type(4))) unsigned int v4u;
typedef __attribute__((ext_vector_type(4))) int v4i;
typedef __attribute__((ext_vector_type(8))) int v8i;

// D(16x16,f32) = A(16x4,f32) * B(4x16,f32) + C
__device__ __forceinline__ v8f wmma4(v2f a, v2f b, v8f c) {
  return __builtin_amdgcn_wmma_f32_16x16x4_f32(
      /*neg_a=*/false, a, /*neg_b=*/false, b,
      /*c_mod=*/(short)0, c, /*reuse_a=*/false, /*reuse_b=*/false);
}

__device__ __forceinline__ float sigmf(float x) { return 1.0f / (1.0f + __expf(-x)); }

// ---------------------------------------------------------------------------
// 0) initial state: h = h0 + 0.1*N(0,1), c = c0 + 0.1*N(0,1) (hash-based RNG)
// ---------------------------------------------------------------------------
__device__ __forceinline__ uint32_t hashu(uint32_t x) {
  x ^= x >> 16; x *= 0x7feb352du;
  x ^= x >> 15; x *= 0x846ca68bu;
  x ^= x >> 16; return x;
}

__global__ void k_init_state(const float* __restrict__ h0, const float* __restrict__ c0,
                             float* __restrict__ h, float* __restrict__ c) {
  int idx = blockIdx.x * blockDim.x + threadIdx.x;
  if (idx >= BATCH * NHID) return;
  int j = idx % NHID;
  uint32_t r0 = hashu(0x9e3779b9u + 4u * (uint32_t)idx);
  uint32_t r1 = hashu(0x85ebca6bu + 4u * (uint32_t)idx);
  uint32_t r2 = hashu(0xc2b2ae35u + 4u * (uint32_t)idx);
  uint32_t r3 = hashu(0x27d4eb2fu + 4u * (uint32_t)idx);
  float u1 = ((float)r0 + 1.0f) * 2.3283064e-10f;
  float u2 = (float)r1 * 2.3283064e-10f;
  float u3 = ((float)r2 + 1.0f) * 2.3283064e-10f;
  float u4 = (float)r3 * 2.3283064e-10f;
  float m1 = __fsqrt_rn(-2.0f * __logf(u1));
  float m2 = __fsqrt_rn(-2.0f * __logf(u3));
  float z0 = m1 * __cosf(6.2831853f * u2);
  float z1 = m2 * __cosf(6.2831853f * u4);
  h[idx] = h0[j] + 0.1f * z0;
  c[idx] = c0[j] + 0.1f * z1;
}

// ---------------------------------------------------------------------------
// 1) C[M,N] = A[M,K] @ W[N,K]^T ; each wave owns a 16x32 strip (A reused x2)
// ---------------------------------------------------------------------------
__global__ void k_gemm_nt(const float* __restrict__ A, const float* __restrict__ W,
                          float* __restrict__ C, int M, int N, int K) {
  const int lane = threadIdx.x & 31;
  const int lr = lane & 15;
  const int lh = lane >> 4;
  const int Nt2 = N >> 5;                       // 32-wide column pairs
  int tp = blockIdx.x * (blockDim.x >> 5) + (threadIdx.x >> 5);
  if (tp >= (M >> 4) * Nt2) return;
  const int m0 = (tp / Nt2) << 4;
  const int n0 = (tp % Nt2) << 5;

  const float* Arow  = A + (size_t)(m0 + lr) * K;
  const float* Wrow0 = W + (size_t)(n0 + lr) * K;
  const float* Wrow1 = W + (size_t)(n0 + 16 + lr) * K;

  v8f acc0 = {0.f, 0.f, 0.f, 0.f, 0.f, 0.f, 0.f, 0.f};
  v8f acc1 = acc0;
  for (int kb = 0; kb < K; kb += 4) {
    v2f a  = *(const v2f*)(Arow + kb + 2 * lh);    // A[m][k..k+1]
    v2f b0 = *(const v2f*)(Wrow0 + kb + 2 * lh);   // Bgemm[k][n]   = W[n][k]
    v2f b1 = *(const v2f*)(Wrow1 + kb + 2 * lh);
    acc0 = wmma4(a, b0, acc0);
    acc1 = wmma4(a, b1, acc1);
  }
#pragma unroll
  for (int r = 0; r < 8; ++r) {
    C[(size_t)(m0 + 8 * lh + r) * N + n0 + lr]      = acc0[r];
    C[(size_t)(m0 + 8 * lh + r) * N + n0 + 16 + lr] = acc1[r];
  }
}

// ---------------------------------------------------------------------------
// 2) per-channel mean over the batch for both gate matrices: mean[2*C4]
// ---------------------------------------------------------------------------
__global__ void k_mean(const float* __restrict__ Y, float* __restrict__ mean) {
  int idx = blockIdx.x * blockDim.x + threadIdx.x;
  if (idx >= 2 * C4) return;
  const float* base = Y + (size_t)(idx / C4) * BATCH * C4 + (idx % C4);
  float s = 0.f;
  for (int b = 0; b < BATCH; ++b) s += base[(size_t)b * C4];
  mean[idx] = s * (1.0f / (float)BATCH);
}

// ---------------------------------------------------------------------------
// 3) Sigma[gg] = Xc @ Xc^T / N + eps*I  (Xc = (Y[:,group]-mean)^T, 160x1024)
//    grid (25, 20), block 128 => 4 tiles/block, 100 tiles/group
// ---------------------------------------------------------------------------
__global__ void k_sigma(const float* __restrict__ Y, const float* __restrict__ mean,
                        float* __restrict__ Sigma) {
  const int lane = threadIdx.x & 31;
  const int lr = lane & 15;
  const int lh = lane >> 4;
  const int gg = blockIdx.y;
  const int mat = gg / NG;
  const int g0 = (gg % NG) * D;
  const int t = blockIdx.x * 4 + (threadIdx.x >> 5);
  const int ti = t / 10, tj = t % 10;

  const float* Ym = Y + (size_t)mat * BATCH * C4;
  const int colA = g0 + ti * 16 + lr;
  const int colB = g0 + tj * 16 + lr;
  const float mA = mean[mat * C4 + colA];
  const float mB = mean[mat * C4 + colB];

  v8f acc = {0.f, 0.f, 0.f, 0.f, 0.f, 0.f, 0.f, 0.f};
  for (int kb = 0; kb < BATCH; kb += 4) {
    const int m = kb + 2 * lh;
    v2f a, b;
    a.x = Ym[(size_t)m * C4 + colA] - mA;
    a.y = Ym[(size_t)(m + 1) * C4 + colA] - mA;
    b.x = Ym[(size_t)m * C4 + colB] - mB;
    b.y = Ym[(size_t)(m + 1) * C4 + colB] - mB;
    acc = wmma4(a, b, acc);
  }
  float* Sg = Sigma + (size_t)gg * D * D;
#pragma unroll
  for (int r = 0; r < 8; ++r) {
    int d = ti * 16 + 8 * lh + r;
    int e = tj * 16 + lr;
    float v = acc[r] * (1.0f / (float)BATCH);
    if (d == e) v += EPSV;
    Sg[d * D + e] = v;
  }
}

// ---------------------------------------------------------------------------
// 4) Fused Newton-Schulz: one WG (32 waves) per group; P, A, B live in LDS
//    (3 x 160x164 f32 = 307.5 KB of the WGP's 320 KB). Sigma stays in L2;
//    P <- 1.5P - 0.5*rTr*(B @ Sigma), wm = P*sqrt(rTr) written at the end.
//    2-row register blocking: each wave owns 32x16 strips (B frag reused x2).
// ---------------------------------------------------------------------------
__global__ void __launch_bounds__(1024) k_ns(const float* __restrict__ Sigma,
                                             float* __restrict__ WM) {
  extern __shared__ float sm[];
  float* P  = sm;
  float* Am = sm + D * LDP;
  float* Bm = sm + 2 * D * LDP;
  __shared__ float s_rtr;

  const int gg = blockIdx.x;
  const float* Sg = Sigma + (size_t)gg * D * D;
  const int tid = threadIdx.x;
  const int wid = tid >> 5;
  const int lane = tid & 31;
  const int lr = lane & 15;
  const int lh = lane >> 4;

  if (tid == 0) {
    float tr = 0.f;
    for (int d = 0; d < D; ++d) tr += Sg[d * D + d];
    s_rtr = 1.0f / tr;
  }
  for (int i = tid; i < D * D; i += 1024) {
    int d = i / D, e = i % D;
    P[d * LDP + e] = (d == e) ? 1.0f : 0.0f;
  }
  __syncthreads();
  const float rtr = s_rtr;

  for (int it = 0; it < T_NS; ++it) {
    // Am = P @ P
    for (int t = wid; t < 50; t += 32) {
      const int i0 = (t / 10) * 32, tj = t % 10;
      v8f acc0 = {0.f, 0.f, 0.f, 0.f, 0.f, 0.f, 0.f, 0.f};
      v8f acc1 = acc0;
      for (int kb = 0; kb < D; kb += 4) {
        v2f a0 = *(const v2f*)(P + (i0 + lr) * LDP + kb + 2 * lh);
        v2f a1 = *(const v2f*)(P + (i0 + 16 + lr) * LDP + kb + 2 * lh);
        v2f b;
        b.x = P[(kb + 2 * lh) * LDP + tj * 16 + lr];
        b.y = P[(kb + 2 * lh + 1) * LDP + tj * 16 + lr];
        acc0 = wmma4(a0, b, acc0);
        acc1 = wmma4(a1, b, acc1);
      }
#pragma unroll
      for (int r = 0; r < 8; ++r) {
        Am[(i0 + 8 * lh + r) * LDP + tj * 16 + lr]      = acc0[r];
        Am[(i0 + 16 + 8 * lh + r) * LDP + tj * 16 + lr] = acc1[r];
      }
    }
    __syncthreads();
    // Bm = Am @ P
    for (int t = wid; t < 50; t += 32) {
      const int i0 = (t / 10) * 32, tj = t % 10;
      v8f acc0 = {0.f, 0.f, 0.f, 0.f, 0.f, 0.f, 0.f, 0.f};
      v8f acc1 = acc0;
      for (int kb = 0; kb < D; kb += 4) {
        v2f a0 = *(const v2f*)(Am + (i0 + lr) * LDP + kb + 2 * lh);
        v2f a1 = *(const v2f*)(Am + (i0 + 16 + lr) * LDP + kb + 2 * lh);
        v2f b;
        b.x = P[(kb + 2 * lh) * LDP + tj * 16 + lr];
        b.y = P[(kb + 2 * lh + 1) * LDP + tj * 16 + lr];
        acc0 = wmma4(a0, b, acc0);
        acc1 = wmma4(a1, b, acc1);
      }
#pragma unroll
      for (int r = 0; r < 8; ++r) {
        Bm[(i0 + 8 * lh + r) * LDP + tj * 16 + lr]      = acc0[r];
        Bm[(i0 + 16 + 8 * lh + r) * LDP + tj * 16 + lr] = acc1[r];
      }
    }
    __syncthreads();
    // P = 1.5P - 0.5*rtr*(Bm @ Sigma)   (each wave touches only its own tiles)
    for (int t = wid; t < 50; t += 32) {
      const int i0 = (t / 10) * 32, tj = t % 10;
      v8f acc0 = {0.f, 0.f, 0.f, 0.f, 0.f, 0.f, 0.f, 0.f};
      v8f acc1 = acc0;
      for (int kb = 0; kb < D; kb += 4) {
        v2f a0 = *(const v2f*)(Bm + (i0 + lr) * LDP + kb + 2 * lh);
        v2f a1 = *(const v2f*)(Bm + (i0 + 16 + lr) * LDP + kb + 2 * lh);
        v2f b;
        b.x = Sg[(kb + 2 * lh) * D + tj * 16 + lr];
        b.y = Sg[(kb + 2 * lh + 1) * D + tj * 16 + lr];
        acc0 = wmma4(a0, b, acc0);
        acc1 = wmma4(a1, b, acc1);
      }
#pragma unroll
      for (int r = 0; r < 8; ++r) {
        int ia = (i0 + 8 * lh + r) * LDP + tj * 16 + lr;
        int ib = (i0 + 16 + 8 * lh + r) * LDP + tj * 16 + lr;
        P[ia] = 1.5f * P[ia] - 0.5f * rtr * acc0[r];
        P[ib] = 1.5f * P[ib] - 0.5f * rtr * acc1[r];
      }
    }
    __syncthreads();
  }

  const float srtr = __fsqrt_rn(rtr);
  float* Wg = WM + (size_t)gg * D * D;
  for (int i = tid; i < D * D; i += 1024)
    Wg[i] = P[(i / D) * LDP + (i % D)] * srtr;
}

// ---------------------------------------------------------------------------
// 5) Whiten: Wout[mat][:,group] = (wm @ Xc)^T
//    grid (4, 20), block 512. wm (160x160 f32, 100 KB) is DMA'd into LDS by
//    the Tensor Data Mover (one tensor_load_to_lds per block, wave 0), then
//    each wave fixes one m-tile, holds 10 accumulators and reuses its Xc
//    B-fragment across 10 WMMAs per k-step.
// ---------------------------------------------------------------------------
__global__ void __launch_bounds__(512) k_whiten(const float* __restrict__ WM,
                                                const float* __restrict__ Y,
                                                const float* __restrict__ mean,
                                                float* __restrict__ Wout) {
  extern __shared__ float sP[];                    // D*D floats
  const int lane = threadIdx.x & 31;
  const int lr = lane & 15;
  const int lh = lane >> 4;
  const int wid = threadIdx.x >> 5;
  const int gg = blockIdx.y;
  const int mat = gg / NG;
  const int g0 = (gg % NG) * D;
  const float* Wg = WM + (size_t)gg * D * D;

  // --- TDM: stage wm[gg] (160x160 f32) global -> LDS, issued by wave 0 ---
  if (threadIdx.x < 32) {
    uint64_t ga = (uint64_t)(uintptr_t)Wg;
    uint32_t lds = (uint32_t)(uintptr_t)sP;
    v4u g0d;
    g0d[0] = 1u;                                            // count=1, user D#
    g0d[1] = lds;                                           // lds_addr (bytes)
    g0d[2] = (uint32_t)ga;                                  // global_addr[31:0]
    g0d[3] = ((uint32_t)(ga >> 32) & 0x01FFFFFFu) | (2u << 30); // ga[56:32] | type=2
    v8i g1d;
    g1d[0] = 0x20000;                                       // data_size=2 (4B), mask=0
    g1d[1] = (int)(((uint32_t)D & 0xFFFFu) << 16);          // tensor_dim0 lo16 @bit48
    g1d[2] = (int)(((uint32_t)D & 0xFFFFu) << 16);          // tensor_dim1 lo16 @bit80
    g1d[3] = (int)((uint32_t)D << 16);                      // tile_dim0 @bit112
    g1d[4] = D;                                             // tile_dim1 ; tile_dim2=0
    g1d[5] = D;                                             // tensor_dim0_stride lo32
    g1d[6] = (int)(((uint32_t)(D * D) & 0xFFFFu) << 16);    // tensor_dim1_stride lo16
    g1d[7] = (int)((uint32_t)(D * D) >> 16);                // tensor_dim1_stride hi
    v4i zd = {0, 0, 0, 0};
    v8i z8 = {0, 0, 0, 0, 0, 0, 0, 0};
    __builtin_amdgcn_tensor_load_to_lds(g0d, g1d, zd, zd, z8, 0);
    __builtin_amdgcn_s_wait_tensorcnt(0);
  }
  __syncthreads();

  const int mt = blockIdx.x * 16 + wid;            // m-tile (0..63)
  const int m0 = mt * 16;
  const float* Ym = Y + (size_t)mat * BATCH * C4 + (size_t)(m0 + lr) * C4 + g0;
  const float* mg = mean + mat * C4 + g0;

  v8f acc[10];
#pragma unroll
  for (int dt = 0; dt < 10; ++dt)
    acc[dt] = (v8f){0.f, 0.f, 0.f, 0.f, 0.f, 0.f, 0.f, 0.f};

  for (int kb = 0; kb < D; kb += 4) {
    v2f y = *(const v2f*)(Ym + kb + 2 * lh);       // Y[m][g0+e..e+1]
    v2f m = *(const v2f*)(mg + kb + 2 * lh);
    v2f b = y - m;                                 // Xc[e][m]
#pragma unroll
    for (int dt = 0; dt < 10; ++dt) {
      v2f a = *(const v2f*)(sP + (dt * 16 + lr) * D + kb + 2 * lh);  // wm[d][e..e+1]
      acc[dt] = wmma4(a, b, acc[dt]);
    }
  }

  float* Wo = Wout + (size_t)mat * BATCH * C4 + (size_t)(m0 + lr) * C4 + g0;
#pragma unroll
  for (int dt = 0; dt < 10; ++dt) {
    float* op = Wo + dt * 16 + 8 * lh;             // 8 consecutive channels
    v4f lo = {acc[dt][0], acc[dt][1], acc[dt][2], acc[dt][3]};
    v4f hi = {acc[dt][4], acc[dt][5], acc[dt][6], acc[dt][7]};
    *(v4f*)op = lo;
    *(v4f*)(op + 4) = hi;
  }
}

// ---------------------------------------------------------------------------
// 6) Gate combine + LSTM cell update (elementwise over BATCH x NHID)
// ---------------------------------------------------------------------------
__global__ void k_cell(const float* __restrict__ Wout,
                       const float* __restrict__ bn_i_w, const float* __restrict__ bn_i_b,
                       const float* __restrict__ bn_h_w, const float* __restrict__ bn_h_b,
                       const float* __restrict__ bias_ih, const float* __restrict__ bias_hh,
                       float* __restrict__ h, float* __restrict__ cst) {
  int idx = blockIdx.x * blockDim.x + threadIdx.x;
  if (idx >= BATCH * NHID) return;
  const int b = idx / NHID;
  const int j = idx % NHID;
  const float* Wx = Wout + (size_t)b * C4;
  const float* Wh = Wout + (size_t)BATCH * C4 + (size_t)b * C4;

  float g4[4];
#pragma unroll
  for (int q = 0; q < 4; ++q) {
    int c = q * NHID + j;
    g4[q] = (Wx[c] * bn_i_w[c] + bn_i_b[c] + bias_ih[c]) +
            (Wh[c] * bn_h_w[c] + bn_h_b[c] + bias_hh[c]);
  }
  float cn = sigmf(g4[1]) * cst[idx] + sigmf(g4[0]) * tanhf(g4[2]);
  float hn = sigmf(g4[3]) * tanhf(cn);
  cst[idx] = cn;
  h[idx] = hn;
}

// ---------------------------------------------------------------------------
// 7) decoder: out = h @ dec_w^T + dec_b   (1024x400 @ 400x10)
// ---------------------------------------------------------------------------
__global__ void k_dec(const float* __restrict__ h, const float* __restrict__ dec_w,
                      const float* __restrict__ dec_b, float* __restrict__ out) {
  int idx = blockIdx.x * blockDim.x + threadIdx.x;
  if (idx >= BATCH * 10) return;
  const int b = idx / 10, k = idx % 10;
  const float* hr = h + (size_t)b * NHID;
  const float* wr = dec_w + (size_t)k * NHID;
  float s = dec_b[k];
  for (int j = 0; j < NHID; ++j) s += hr[j] * wr[j];
  out[idx] = s;
}

// ---------------------------------------------------------------------------
// host-side launch
// ---------------------------------------------------------------------------
extern "C" void kernel_launch(void* const* d_in, const int* in_sizes, int n_in,
                              void* d_out, int out_size, void* d_ws, size_t ws_size,
                              hipStream_t stream) {
  const float* input     = (const float*)d_in[0];   // (SEQ, B, NINP)
  const float* weight_ih = (const float*)d_in[1];   // (C4, NINP)
  const float* weight_hh = (const float*)d_in[2];   // (C4, NHID)
  const float* bias_ih   = (const float*)d_in[3];
  const float* bias_hh   = (const float*)d_in[4];
  const float* bn_i_w    = (const float*)d_in[5];
  const float* bn_i_b    = (const float*)d_in[6];
  const float* bn_h_w    = (const float*)d_in[7];
  const float* bn_h_b    = (const float*)d_in[8];
  const float* h0        = (const float*)d_in[9];
  const float* c0        = (const float*)d_in[10];
  const float* dec_w     = (const float*)d_in[11];
  const float* dec_b     = (const float*)d_in[12];
  float* out = (float*)d_out;

  // workspace layout (floats)
  float* ws    = (float*)d_ws;
  float* GY    = ws;                                  // 2 * B * C4 raw gate pre-acts
  float* WOUT  = GY   + (size_t)2 * BATCH * C4;       // 2 * B * C4 whitened
  float* meanv = WOUT + (size_t)2 * BATCH * C4;       // 2 * C4
  float* Sigma = meanv + 2 * C4;                      // 20 * 160 * 160
  float* WM    = Sigma + (size_t)NG2 * D * D;         // 20 * 160 * 160
  float* h     = WM    + (size_t)NG2 * D * D;         // B * NHID
  float* c     = h     + (size_t)BATCH * NHID;        // B * NHID

  const size_t nsLds = (size_t)3 * D * LDP * sizeof(float);   // 314,880 B
  const size_t whLds = (size_t)D * D * sizeof(float);         // 102,400 B

  k_init_state<<<(BATCH * NHID + 255) / 256, 256, 0, stream>>>(h0, c0, h, c);

  for (int t = 0; t < SEQ; ++t) {
    const float* x_t = input + (size_t)t * BATCH * NINP;
    // raw gate GEMMs: 3200 16x32 strips each, 8 waves/block
    k_gemm_nt<<<400, 256, 0, stream>>>(x_t, weight_ih, GY, BATCH, C4, NINP);
    k_gemm_nt<<<400, 256, 0, stream>>>(h,   weight_hh, GY + (size_t)BATCH * C4,
                                       BATCH, C4, NHID);
    k_mean<<<(2 * C4 + 255) / 256, 256, 0, stream>>>(GY, meanv);
    k_sigma<<<dim3(25, NG2), 128, 0, stream>>>(GY, meanv, Sigma);
    k_ns<<<NG2, 1024, nsLds, stream>>>(Sigma, WM);
    k_whiten<<<dim3(4, NG2), 512, whLds, stream>>>(WM, GY, meanv, WOUT);
    k_cell<<<(BATCH * NHID + 255) / 256, 256, 0, stream>>>(
        WOUT, bn_i_w, bn_i_b, bn_h_w, bn_h_b, bias_ih, bias_hh, h, c);
  }

  k_dec<<<(BATCH * 10 + 255) / 256, 256, 0, stream>>>(h, dec_w, dec_b, out);
  (void)in_sizes; (void)n_in; (void)out_size; (void)ws_size;
}